// VerticalLinePool_25623774888171
// MI455X (gfx1250) — compile-verified
//
#include <hip/hip_runtime.h>
#include <stdint.h>

// Problem geometry (from reference: x = (8, 128, 256, 256) f32, scan along H).
#define HDIM 256
#define WDIM 256
#define CH 32                    // rows per staged chunk
#define NCHUNK (HDIM / CH)       // 8 chunks per plane
#define CHUNK_ELEMS (CH * WDIM)  // 8192 f32 = 32 KB

typedef __attribute__((ext_vector_type(4))) unsigned int u32x4;
typedef __attribute__((ext_vector_type(8))) int i32x8;
typedef __attribute__((ext_vector_type(4))) int i32x4;

#if __has_builtin(__builtin_amdgcn_tensor_load_to_lds) && __has_builtin(__builtin_amdgcn_s_wait_tensorcnt)
#define USE_TDM 1
#else
#define USE_TDM 0
#endif

#if USE_TDM
// Issue one TDM DMA: CHUNK_ELEMS contiguous f32 from global -> LDS.
// D# per CDNA5 ISA. 1-D tile (tensor_dim1 = 1).
__device__ __forceinline__ void tdm_load_chunk(unsigned lds_byte_off,
                                               unsigned long long gaddr) {
  u32x4 g0;
  g0.x = 1u;                                                  // count=1, user D#
  g0.y = lds_byte_off;                                        // lds_addr (bytes)
  g0.z = (unsigned)(gaddr & 0xFFFFFFFFull);                   // global_addr[31:0]
  g0.w = (unsigned)((gaddr >> 32) & 0x01FFFFFFull)            // global_addr[56:32]
         | 0x80000000u;                                       // type = 2 ("image")

  i32x8 g1;
  g1[0] = 0x00020000;                        // data_size = 2 (4 bytes), no multicast/pad/iter
  g1[1] = (int)(CHUNK_ELEMS << 16);          // tensor_dim0[15:0] in bits 63:48
  g1[2] = (int)(1u << 16);                   // tensor_dim0 hi = 0 | tensor_dim1 = 1
  g1[3] = (int)(CHUNK_ELEMS << 16);          // tensor_dim1 hi = 0 | tile_dim0
  g1[4] = 0;                                 // tile_dim1 = 0 (unused), tile_dim2 = 0
  g1[5] = CHUNK_ELEMS;                       // tensor_dim0_stride[31:0]
  g1[6] = 0;                                 // stride hi | tensor_dim1_stride lo
  g1[7] = 0;

  i32x4 gz = {0, 0, 0, 0};
#if defined(__clang_major__) && (__clang_major__ >= 23)
  i32x8 gz8 = {0, 0, 0, 0, 0, 0, 0, 0};
  __builtin_amdgcn_tensor_load_to_lds(g0, g1, gz, gz, gz8, 0);
#else
  __builtin_amdgcn_tensor_load_to_lds(g0, g1, gz, gz, 0);
#endif
}
#endif

__global__ __launch_bounds__(256) void VerticalLinePool_kernel(
    const float* __restrict__ x, float* __restrict__ out) {
  const int plane = blockIdx.x;                       // (n,c) plane index
  const size_t base = (size_t)plane * (HDIM * WDIM);  // plane is 256 KB contiguous
  const int w = threadIdx.x;                          // one W column per thread
  float running = -__builtin_inff();

#if USE_TDM
  __shared__ float lds[2][CHUNK_ELEMS];               // 64 KB, double buffered

  // Scalar-uniform wave-0 guard: TDM ignores EXEC, so the skip must be a
  // true scalar branch (readfirstlane -> SGPR compare -> s_cbranch).
  const bool wave0 =
      (__builtin_amdgcn_readfirstlane((int)(threadIdx.x >> 5)) == 0);

  // Prologue: stage the bottom chunk (scan is bottom-to-top).
  if (wave0) {
    unsigned off0 = (unsigned)(unsigned long long)(uintptr_t)&lds[0][0];
    unsigned long long g = (unsigned long long)(uintptr_t)(
        x + base + (size_t)(NCHUNK - 1) * CHUNK_ELEMS);
    tdm_load_chunk(off0, g);
    __builtin_amdgcn_s_wait_tensorcnt(0);
  }
  __syncthreads();

  int buf = 0;
  for (int c = NCHUNK - 1; c >= 0; --c) {
    // Kick off the next chunk's DMA into the other buffer (overlaps compute).
    if (c > 0 && wave0) {
      unsigned off = (unsigned)(unsigned long long)(uintptr_t)&lds[buf ^ 1][0];
      unsigned long long g = (unsigned long long)(uintptr_t)(
          x + base + (size_t)(c - 1) * CHUNK_ELEMS);
      tdm_load_chunk(off, g);
    }

    const float* src = &lds[buf][0];
    const size_t obase = base + (size_t)c * CHUNK_ELEMS;
#pragma unroll 8
    for (int r = CH - 1; r >= 0; --r) {
      float v = src[r * WDIM + w];              // conflict-free: 32 lanes -> 32 banks
      running = fmaxf(running, v);              // the only serial dependence
      __builtin_nontemporal_store(running, &out[obase + (size_t)r * WDIM + w]);
    }

    if (wave0) __builtin_amdgcn_s_wait_tensorcnt(0);
    __syncthreads();                            // buffer (buf^1) now valid for all
    buf ^= 1;
  }
#else
  // Fallback (no TDM builtin on this toolchain): plain coalesced streaming.
  const float* col = x + base + w;
  float* ocol = out + base + w;
#pragma unroll 8
  for (int h = HDIM - 1; h >= 0; --h) {
    running = fmaxf(running, col[(size_t)h * WDIM]);
    __builtin_nontemporal_store(running, &ocol[(size_t)h * WDIM]);
  }
#endif
}

extern "C" void kernel_launch(void* const* d_in, const int* in_sizes, int n_in,
                              void* d_out, int out_size, void* d_ws, size_t ws_size,
                              hipStream_t stream) {
  const float* x = (const float*)d_in[0];
  float* out = (float*)d_out;
  const int planes = in_sizes[0] / (HDIM * WDIM);  // 8*128 = 1024 blocks
  VerticalLinePool_kernel<<<planes, 256, 0, stream>>>(x, out);
}